// GAT_15221364097250
// MI455X (gfx1250) — compile-verified
//
#include <hip/hip_runtime.h>
#include <math.h>

#define N_NODES   50000
#define N_EDGES   1600000
#define F_IN      512
#define H1        8
#define C1        8
#define F1        64          // H1*C1
#define NCLS      40
#define NEG_SLOPE 0.2f
#define EP        (N_EDGES + N_NODES)   // edges + self loops
#define MTILES    (N_NODES / 16)        // 3125, exact

typedef float v2f __attribute__((ext_vector_type(2)));
typedef float v8f __attribute__((ext_vector_type(8)));

// ---------------------------------------------------------------------------
// float atomic max via monotone int mapping (m initialized to -inf)
// ---------------------------------------------------------------------------
__device__ __forceinline__ void atomicMaxF(float* addr, float v) {
    if (v >= 0.0f) atomicMax((int*)addr, __float_as_int(v));
    else           atomicMin((unsigned int*)addr, (unsigned int)__float_as_int(v));
}

// ---------------------------------------------------------------------------
// workspace init: m = -inf, s = 0, agg = 0
// ---------------------------------------------------------------------------
__global__ void init_ws_k(float* m1, float* s1, float* agg1,
                          float* m2, float* s2, float* agg2) {
    int i = blockIdx.x * blockDim.x + threadIdx.x;
    if (i < N_NODES * F1)  agg1[i] = 0.0f;
    if (i < N_NODES * NCLS) agg2[i] = 0.0f;
    if (i < N_NODES * H1)  { m1[i] = -INFINITY; s1[i] = 0.0f; }
    if (i < N_NODES)       { m2[i] = -INFINITY; s2[i] = 0.0f; }
}

// ---------------------------------------------------------------------------
// GEMM1: h1[N,64] = x[N,512] @ w1[512,64]   (f32 WMMA 16x16x4)
// block = 128 threads = 4 waves; wave w handles M-tile blockIdx*4+w (16 rows),
// all 4 N-tiles. W1 staged in LDS in 128-K chunks, k-pair interleaved:
//   lds[(k>>1)*128 + col*2 + (k&1)]   -> float2 (k, k+1) per col, 8B aligned
// ---------------------------------------------------------------------------
__global__ __launch_bounds__(128) void gemm1_wmma_k(const float* __restrict__ x,
                                                    const float* __restrict__ w1,
                                                    float* __restrict__ h1) {
    __shared__ float lds[128 * 64];
    const int tid  = threadIdx.x;
    const int wave = tid >> 5;
    const int lane = tid & 31;
    const int mt   = blockIdx.x * 4 + wave;
    const bool active = (mt < MTILES);

    const int row16 = lane & 15;
    const int khalf = (lane >> 4) * 2;          // 0 or 2

    v8f acc0 = {}, acc1 = {}, acc2 = {}, acc3 = {};
    const float* xr = x + (size_t)((active ? mt : 0) * 16 + row16) * F_IN + khalf;

    for (int kc = 0; kc < F_IN; kc += 128) {
        // cooperative load of 128x64 chunk of w1 into pair-interleaved LDS
        const float4* src4 = (const float4*)(w1 + (size_t)kc * F1);
        for (int i = tid; i < (128 * 64) / 4; i += 128) {
            float4 v = src4[i];
            int base = i * 4;
            int k = base >> 6;          // row within chunk
            int c = base & 63;          // col
            float* o = lds + (k >> 1) * 128 + (k & 1);
            o[(c + 0) * 2] = v.x;
            o[(c + 1) * 2] = v.y;
            o[(c + 2) * 2] = v.z;
            o[(c + 3) * 2] = v.w;
        }
        __syncthreads();

        if (active) {
            for (int k0 = 0; k0 < 128; k0 += 4) {
                v2f a = *(const v2f*)(xr + kc + k0);
                const float* bb = lds + ((k0 + khalf) >> 1) * 128 + row16 * 2;
                v2f b0 = *(const v2f*)(bb + 0);
                v2f b1 = *(const v2f*)(bb + 32);
                v2f b2 = *(const v2f*)(bb + 64);
                v2f b3 = *(const v2f*)(bb + 96);
                acc0 = __builtin_amdgcn_wmma_f32_16x16x4_f32(false, a, false, b0, (short)0, acc0, false, false);
                acc1 = __builtin_amdgcn_wmma_f32_16x16x4_f32(false, a, false, b1, (short)0, acc1, false, false);
                acc2 = __builtin_amdgcn_wmma_f32_16x16x4_f32(false, a, false, b2, (short)0, acc2, false, false);
                acc3 = __builtin_amdgcn_wmma_f32_16x16x4_f32(false, a, false, b3, (short)0, acc3, false, false);
            }
        }
        __syncthreads();
    }

    if (active) {
        float* out = h1 + (size_t)(mt * 16) * F1;
        const int cl = lane & 15;
        const int mh = (lane >> 4) * 8;
        for (int j = 0; j < 8; ++j) {
            int m = mh + j;
            out[m * F1 +  0 + cl] = acc0[j];
            out[m * F1 + 16 + cl] = acc1[j];
            out[m * F1 + 32 + cl] = acc2[j];
            out[m * F1 + 48 + cl] = acc3[j];
        }
    }
}

// ---------------------------------------------------------------------------
// GEMM2: h2[N,40] = relu(agg1[N,64] + b1) @ w2[64,40]   (f32 WMMA, N padded 48)
// ---------------------------------------------------------------------------
__global__ __launch_bounds__(128) void gemm2_wmma_k(const float* __restrict__ agg1,
                                                    const float* __restrict__ b1,
                                                    const float* __restrict__ w2,
                                                    float* __restrict__ h2) {
    __shared__ float lds[64 * 48 + 64];     // padded w2 (pair layout) + b1
    float* ldsb = lds + 64 * 48;
    const int tid  = threadIdx.x;
    const int wave = tid >> 5;
    const int lane = tid & 31;

    for (int i = tid; i < 64 * 48; i += 128) {
        int k = i / 48, c = i - k * 48;
        float v = (c < NCLS) ? w2[k * NCLS + c] : 0.0f;
        lds[(k >> 1) * 96 + c * 2 + (k & 1)] = v;
    }
    if (tid < 64) ldsb[tid] = b1[tid];
    __syncthreads();

    const int mt = blockIdx.x * 4 + wave;
    if (mt >= MTILES) return;

    const int row16 = lane & 15;
    const int khalf = (lane >> 4) * 2;
    const int row   = mt * 16 + row16;

    v8f acc0 = {}, acc1 = {}, acc2 = {};
    for (int k0 = 0; k0 < F1; k0 += 4) {
        int kk = k0 + khalf;
        v2f av = *(const v2f*)(agg1 + (size_t)row * F1 + kk);
        v2f bv = *(const v2f*)(ldsb + kk);
        v2f a;
        a.x = fmaxf(av.x + bv.x, 0.0f);
        a.y = fmaxf(av.y + bv.y, 0.0f);
        const float* bb = lds + (kk >> 1) * 96 + row16 * 2;
        v2f b0 = *(const v2f*)(bb + 0);
        v2f b1v = *(const v2f*)(bb + 32);
        v2f b2v = *(const v2f*)(bb + 64);
        acc0 = __builtin_amdgcn_wmma_f32_16x16x4_f32(false, a, false, b0,  (short)0, acc0, false, false);
        acc1 = __builtin_amdgcn_wmma_f32_16x16x4_f32(false, a, false, b1v, (short)0, acc1, false, false);
        acc2 = __builtin_amdgcn_wmma_f32_16x16x4_f32(false, a, false, b2v, (short)0, acc2, false, false);
    }

    const int cl = lane & 15;
    const int mh = (lane >> 4) * 8;
    for (int j = 0; j < 8; ++j) {
        int r = mt * 16 + mh + j;
        h2[(size_t)r * NCLS +  0 + cl] = acc0[j];
        h2[(size_t)r * NCLS + 16 + cl] = acc1[j];
        if (cl < 8) h2[(size_t)r * NCLS + 32 + cl] = acc2[j];
    }
}

// ---------------------------------------------------------------------------
// attention coefficients
// ---------------------------------------------------------------------------
__global__ void att1_k(const float* __restrict__ h1,
                       const float* __restrict__ att_src, const float* __restrict__ att_dst,
                       float* __restrict__ asrc, float* __restrict__ adst) {
    int i = blockIdx.x * blockDim.x + threadIdx.x;   // node*8 + head
    if (i >= N_NODES * H1) return;
    int node = i >> 3, h = i & 7;
    const float* hp = h1 + (size_t)node * F1 + h * C1;
    const float* as = att_src + h * C1;
    const float* ad = att_dst + h * C1;
    float s = 0.0f, d = 0.0f;
    for (int c = 0; c < C1; ++c) { float v = hp[c]; s += v * as[c]; d += v * ad[c]; }
    asrc[i] = s; adst[i] = d;
}

__global__ void att2_k(const float* __restrict__ h2,
                       const float* __restrict__ att_src, const float* __restrict__ att_dst,
                       float* __restrict__ asrc, float* __restrict__ adst) {
    int n = blockIdx.x * blockDim.x + threadIdx.x;
    if (n >= N_NODES) return;
    const float* hp = h2 + (size_t)n * NCLS;
    float s = 0.0f, d = 0.0f;
    for (int c = 0; c < NCLS; ++c) { float v = hp[c]; s += v * att_src[c]; d += v * att_dst[c]; }
    asrc[n] = s; adst[n] = d;
}

// ---------------------------------------------------------------------------
// edge helpers
// ---------------------------------------------------------------------------
__device__ __forceinline__ void edge_sd(const long long* __restrict__ ei, int e, int& s, int& d) {
    if (e < N_EDGES) { s = (int)ei[e]; d = (int)ei[N_EDGES + e]; }
    else             { s = d = e - N_EDGES; }           // self loop
}

// layer1 pass1: segment max
__global__ void edge1_max_k(const long long* __restrict__ ei,
                            const float* __restrict__ asrc, const float* __restrict__ adst,
                            float* __restrict__ m) {
    unsigned t = blockIdx.x * blockDim.x + threadIdx.x;
    if (t >= (unsigned)EP * H1) return;
    int e = t >> 3, h = t & 7, s, d;
    edge_sd(ei, e, s, d);
    float v = asrc[s * H1 + h] + adst[d * H1 + h];
    v = v > 0.0f ? v : NEG_SLOPE * v;
    atomicMaxF(&m[d * H1 + h], v);
}

// layer1 pass2: exp-sum
__global__ void edge1_sum_k(const long long* __restrict__ ei,
                            const float* __restrict__ asrc, const float* __restrict__ adst,
                            const float* __restrict__ m, float* __restrict__ sb) {
    unsigned t = blockIdx.x * blockDim.x + threadIdx.x;
    if (t >= (unsigned)EP * H1) return;
    int e = t >> 3, h = t & 7, s, d;
    edge_sd(ei, e, s, d);
    float v = asrc[s * H1 + h] + adst[d * H1 + h];
    v = v > 0.0f ? v : NEG_SLOPE * v;
    atomicAdd(&sb[d * H1 + h], expf(v - m[d * H1 + h]));
}

// layer1 pass3: weighted aggregation (thread per edge-channel)
__global__ void edge1_agg_k(const long long* __restrict__ ei,
                            const float* __restrict__ asrc, const float* __restrict__ adst,
                            const float* __restrict__ m, const float* __restrict__ sb,
                            const float* __restrict__ h1, float* __restrict__ agg) {
    unsigned t = blockIdx.x * blockDim.x + threadIdx.x;
    if (t >= (unsigned)EP * F1) return;
    int e = t >> 6, c = t & 63, h = c >> 3, s, d;
    edge_sd(ei, e, s, d);
    float v = asrc[s * H1 + h] + adst[d * H1 + h];
    v = v > 0.0f ? v : NEG_SLOPE * v;
    float alpha = expf(v - m[d * H1 + h]) / (sb[d * H1 + h] + 1e-16f);
    atomicAdd(&agg[(size_t)d * F1 + c], alpha * h1[(size_t)s * F1 + c]);
}

// layer2 pass1
__global__ void edge2_max_k(const long long* __restrict__ ei,
                            const float* __restrict__ asrc, const float* __restrict__ adst,
                            float* __restrict__ m) {
    unsigned e = blockIdx.x * blockDim.x + threadIdx.x;
    if (e >= (unsigned)EP) return;
    int s, d; edge_sd(ei, (int)e, s, d);
    float v = asrc[s] + adst[d];
    v = v > 0.0f ? v : NEG_SLOPE * v;
    atomicMaxF(&m[d], v);
}

// layer2 pass2
__global__ void edge2_sum_k(const long long* __restrict__ ei,
                            const float* __restrict__ asrc, const float* __restrict__ adst,
                            const float* __restrict__ m, float* __restrict__ sb) {
    unsigned e = blockIdx.x * blockDim.x + threadIdx.x;
    if (e >= (unsigned)EP) return;
    int s, d; edge_sd(ei, (int)e, s, d);
    float v = asrc[s] + adst[d];
    v = v > 0.0f ? v : NEG_SLOPE * v;
    atomicAdd(&sb[d], expf(v - m[d]));
}

// layer2 pass3 (64-padded channel index, skip c >= 40 for coalescing)
__global__ void edge2_agg_k(const long long* __restrict__ ei,
                            const float* __restrict__ asrc, const float* __restrict__ adst,
                            const float* __restrict__ m, const float* __restrict__ sb,
                            const float* __restrict__ h2, float* __restrict__ agg) {
    unsigned t = blockIdx.x * blockDim.x + threadIdx.x;
    if (t >= (unsigned)EP * 64) return;
    int e = t >> 6, c = t & 63;
    if (c >= NCLS) return;
    int s, d; edge_sd(ei, e, s, d);
    float v = asrc[s] + adst[d];
    v = v > 0.0f ? v : NEG_SLOPE * v;
    float alpha = expf(v - m[d]) / (sb[d] + 1e-16f);
    atomicAdd(&agg[(size_t)d * NCLS + c], alpha * h2[(size_t)s * NCLS + c]);
}

// ---------------------------------------------------------------------------
// final: out = log_softmax(agg2 + b2)
// ---------------------------------------------------------------------------
__global__ void final_k(const float* __restrict__ agg2, const float* __restrict__ b2,
                        float* __restrict__ out) {
    int n = blockIdx.x * blockDim.x + threadIdx.x;
    if (n >= N_NODES) return;
    const float* r = agg2 + (size_t)n * NCLS;
    float v[NCLS];
    float mx = -INFINITY;
    for (int c = 0; c < NCLS; ++c) { v[c] = r[c] + b2[c]; mx = fmaxf(mx, v[c]); }
    float s = 0.0f;
    for (int c = 0; c < NCLS; ++c) s += expf(v[c] - mx);
    float ls = logf(s) + mx;
    float* o = out + (size_t)n * NCLS;
    for (int c = 0; c < NCLS; ++c) o[c] = v[c] - ls;
}

// ---------------------------------------------------------------------------
extern "C" void kernel_launch(void* const* d_in, const int* in_sizes, int n_in,
                              void* d_out, int out_size, void* d_ws, size_t ws_size,
                              hipStream_t stream) {
    const float*     x        = (const float*)d_in[0];
    const long long* ei       = (const long long*)d_in[1];
    const float*     w1       = (const float*)d_in[2];
    const float*     att_src1 = (const float*)d_in[3];
    const float*     att_dst1 = (const float*)d_in[4];
    const float*     b1       = (const float*)d_in[5];
    const float*     w2       = (const float*)d_in[6];
    const float*     att_src2 = (const float*)d_in[7];
    const float*     att_dst2 = (const float*)d_in[8];
    const float*     b2       = (const float*)d_in[9];
    float* out = (float*)d_out;

    float* p = (float*)d_ws;
    float* h1    = p; p += (size_t)N_NODES * F1;
    float* asrc1 = p; p += (size_t)N_NODES * H1;
    float* adst1 = p; p += (size_t)N_NODES * H1;
    float* m1    = p; p += (size_t)N_NODES * H1;
    float* s1    = p; p += (size_t)N_NODES * H1;
    float* agg1  = p; p += (size_t)N_NODES * F1;
    float* h2    = p; p += (size_t)N_NODES * NCLS;
    float* asrc2 = p; p += (size_t)N_NODES;
    float* adst2 = p; p += (size_t)N_NODES;
    float* m2    = p; p += (size_t)N_NODES;
    float* s2    = p; p += (size_t)N_NODES;
    float* agg2  = p; p += (size_t)N_NODES * NCLS;

    const int TB = 256;
    auto cdiv = [](long long a, long long b) { return (unsigned)((a + b - 1) / b); };

    init_ws_k<<<cdiv((long long)N_NODES * F1, TB), TB, 0, stream>>>(m1, s1, agg1, m2, s2, agg2);

    // layer 1
    gemm1_wmma_k<<<cdiv(MTILES, 4), 128, 0, stream>>>(x, w1, h1);
    att1_k<<<cdiv((long long)N_NODES * H1, TB), TB, 0, stream>>>(h1, att_src1, att_dst1, asrc1, adst1);
    edge1_max_k<<<cdiv((long long)EP * H1, TB), TB, 0, stream>>>(ei, asrc1, adst1, m1);
    edge1_sum_k<<<cdiv((long long)EP * H1, TB), TB, 0, stream>>>(ei, asrc1, adst1, m1, s1);
    edge1_agg_k<<<cdiv((long long)EP * F1, TB), TB, 0, stream>>>(ei, asrc1, adst1, m1, s1, h1, agg1);

    // layer 2 (relu(agg1+b1) fused into GEMM2 A-load)
    gemm2_wmma_k<<<cdiv(MTILES, 4), 128, 0, stream>>>(agg1, b1, w2, h2);
    att2_k<<<cdiv(N_NODES, TB), TB, 0, stream>>>(h2, att_src2, att_dst2, asrc2, adst2);
    edge2_max_k<<<cdiv(EP, TB), TB, 0, stream>>>(ei, asrc2, adst2, m2);
    edge2_sum_k<<<cdiv(EP, TB), TB, 0, stream>>>(ei, asrc2, adst2, m2, s2);
    edge2_agg_k<<<cdiv((long long)EP * 64, TB), TB, 0, stream>>>(ei, asrc2, adst2, m2, s2, h2, agg2);

    final_k<<<cdiv(N_NODES, TB), TB, 0, stream>>>(agg2, b2, out);
}